// SpatialTransformer_89000312308015
// MI455X (gfx1250) — compile-verified
//
#include <hip/hip_runtime.h>

// Spatial transformer (trilinear gather-interp) for MI455X / gfx1250.
// vol: [2,160,192,160,1] f32, trf: [2,160,192,160,3] f32 abs coords.
// Memory-bound: trf/out streamed with NT hints (protect 39MB vol working set
// in the 192MB L2), vol corners fetched as b64 x-pairs (4 gathers/pt not 8).

#define DD 160
#define HH 192
#define WW 160
#define BB 2
#define DHW (DD * HH * WW)        // 4,915,200
#define NTOT (BB * DHW)           // 9,830,400

typedef float f32x2 __attribute__((ext_vector_type(2), aligned(4)));

__device__ __forceinline__ void axis_weights(float loc, float maxf,
                                             int& i0, int& i1,
                                             float& w0, float& w1) {
    // Mirrors reference exactly: clipped coord, clamped floor, clamped floor+1,
    // w0 = l1 - clipped (weight of corner bit 0), w1 = 1 - w0.
    float f  = floorf(loc);
    float cl = fminf(fmaxf(loc, 0.0f), maxf);
    float l0 = fminf(fmaxf(f,   0.0f), maxf);
    float l1 = fminf(l0 + 1.0f, maxf);
    i0 = (int)l0;
    i1 = (int)l1;
    w0 = l1 - cl;
    w1 = 1.0f - w0;
}

__global__ __launch_bounds__(256)
void SpatialTransformer_trilerp_kernel(const float* __restrict__ vol,
                                       const float* __restrict__ trf,
                                       float* __restrict__ out) {
    int idx = blockIdx.x * blockDim.x + threadIdx.x;
    if (idx >= NTOT) return;

    const float* t = trf + (size_t)idx * 3;

    // Run the trf stream ahead of the wave (speculative prefetch: OOB at the
    // tail is silently dropped). Emits global_prefetch_b8 on gfx1250.
    __builtin_prefetch(t + 3 * 4096, 0, 0);

    // Streaming, read-once: non-temporal so 118MB of trf doesn't evict vol in L2.
    float zl = __builtin_nontemporal_load(t + 0);   // dim0 -> D
    float yl = __builtin_nontemporal_load(t + 1);   // dim1 -> H
    float xl = __builtin_nontemporal_load(t + 2);   // dim2 -> W (innermost)

    int z0, z1, y0, y1, x0, x1;
    float wz0, wz1, wy0, wy1, wx0, wx1;
    axis_weights(zl, (float)(DD - 1), z0, z1, wz0, wz1);
    axis_weights(yl, (float)(HH - 1), y0, y1, wy0, wy1);
    axis_weights(xl, (float)(WW - 1), x0, x1, wx0, wx1);
    (void)x1;

    // x-pair gather: corners (x0, x0+1) are 8 contiguous bytes. Base the b64
    // load at px = min(x0, W-2) so it never crosses the row end; when x0 was
    // clamped to W-1 (px+1 == x0 == x1), fold the lo weight into the hi lane.
    int  px      = (x0 < (WW - 2)) ? x0 : (WW - 2);
    bool hiclamp = (x0 > px);
    float a1 = hiclamp ? (wx0 + wx1) : wx1;
    float a0 = hiclamp ? 0.0f        : wx0;

    const float* vb = vol + (idx >= DHW ? (size_t)DHW : (size_t)0);

    int rz0 = z0 * HH, rz1 = z1 * HH;
    int o00 = (rz0 + y0) * WW + px;
    int o01 = (rz0 + y1) * WW + px;
    int o10 = (rz1 + y0) * WW + px;
    int o11 = (rz1 + y1) * WW + px;

    // 4 x b64 gathers (L2-resident after warmup), regular-temporal policy.
    f32x2 p00 = *(const f32x2*)(vb + o00);
    f32x2 p01 = *(const f32x2*)(vb + o01);
    f32x2 p10 = *(const f32x2*)(vb + o10);
    f32x2 p11 = *(const f32x2*)(vb + o11);

    // lerp x, then y, then z (same weights/products as the reference's 8-corner sum)
    float v00 = fmaf(a0, p00.x, a1 * p00.y);
    float v01 = fmaf(a0, p01.x, a1 * p01.y);
    float v10 = fmaf(a0, p10.x, a1 * p10.y);
    float v11 = fmaf(a0, p11.x, a1 * p11.y);

    float vz0 = fmaf(wy0, v00, wy1 * v01);
    float vz1 = fmaf(wy0, v10, wy1 * v11);
    float res = fmaf(wz0, vz0, wz1 * vz1);

    // Streaming write-once output: non-temporal store.
    __builtin_nontemporal_store(res, out + idx);
}

extern "C" void kernel_launch(void* const* d_in, const int* in_sizes, int n_in,
                              void* d_out, int out_size, void* d_ws, size_t ws_size,
                              hipStream_t stream) {
    const float* vol = (const float*)d_in[0];   // [2,160,192,160,1] f32
    const float* trf = (const float*)d_in[1];   // [2,160,192,160,3] f32
    float* out = (float*)d_out;                 // [2,160,192,160,1] f32

    const int threads = 256;                    // 8 wave32 per block
    const int blocks  = (NTOT + threads - 1) / threads;   // 38,400 exact
    SpatialTransformer_trilerp_kernel<<<blocks, threads, 0, stream>>>(vol, trf, out);
}